// WeightedThresholdGate_53085795778563
// MI455X (gfx1250) — compile-verified
//
#include <hip/hip_runtime.h>
#include <hip/hip_bf16.h>
#include <cstdint>

// WeightedThresholdGate for MI455X (gfx1250).
// out[b,o] = sigmoid( (softplus(s_raw[o])+1e-6) * (sum_f x[b, idx[o,f]]*w[o,f] - theta[o]) )
//
// Memory-bound (~34 MB HBM floor -> ~1.5us @ 23.3 TB/s). Strategy:
//   * stage 4 rows of x (64 KB) per block into LDS via CDNA5 async loads
//     (global_load_async_to_lds_b128, ASYNCcnt, s_wait_asynccnt)
//   * resolve the FAN_IN=8 random gathers from LDS (64 banks)
//   * software-pipeline the per-output parameter stream (idx/w/theta/s_raw,
//     L2-resident) against the gather+epilogue compute; prefetch 2 tiles ahead.

#define B_DIM   1024
#define IN_DIM  4096
#define OUT_DIM 4096
#define FAN_IN  8

#define ROWS    4                       // batch rows staged in LDS per block (64 KB)
#define OTILE   64                      // outputs per inner tile
#define OSPLIT  4                       // blocks splitting the output dimension
#define THREADS 256                     // 8 wave32
#define OUT_PER_BLOCK (OUT_DIM / OSPLIT)

struct OutParams {
    int4   i0, i1;
    float4 w0, w1;
    float  th, sr;
};

__device__ __forceinline__ OutParams load_params(const int*   __restrict__ idx,
                                                 const float* __restrict__ w,
                                                 const float* __restrict__ theta,
                                                 const float* __restrict__ s_raw,
                                                 int o) {
    OutParams p;
    p.i0 = *(const int4*)  (idx + (size_t)o * FAN_IN);
    p.i1 = *(const int4*)  (idx + (size_t)o * FAN_IN + 4);
    p.w0 = *(const float4*)(w   + (size_t)o * FAN_IN);
    p.w1 = *(const float4*)(w   + (size_t)o * FAN_IN + 4);
    p.th = theta[o];
    p.sr = s_raw[o];
    return p;
}

__global__ __launch_bounds__(THREADS)
void wtg_kernel(const float* __restrict__ x,
                const int*   __restrict__ idx,
                const float* __restrict__ w,
                const float* __restrict__ theta,
                const float* __restrict__ s_raw,
                float*       __restrict__ out)
{
    __shared__ float tile[ROWS * IN_DIM];          // 64 KB of the 320 KB WGP LDS

    const int tid  = threadIdx.x;
    const int row0 = blockIdx.y * ROWS;

    // ---- Kick off async staging of ROWS rows of x into LDS (ASYNCcnt) ----
    {
        const float* src = x + (size_t)row0 * IN_DIM;
        #pragma unroll
        for (int k = 0; k < (ROWS * IN_DIM / 4) / THREADS; ++k) {   // 16 iters
            const int j = tid + k * THREADS;                        // float4 index
            const uint64_t gaddr = (uint64_t)(uintptr_t)(src + (size_t)j * 4);
            const uint32_t laddr = (uint32_t)(uintptr_t)(const void*)&tile[j * 4];
            // VDST = per-lane LDS byte address, VADDR = 64-bit global address
            asm volatile("global_load_async_to_lds_b128 %0, %1, off"
                         :: "v"(laddr), "v"(gaddr) : "memory");
        }
    }

    const int    o_local = tid & (OTILE - 1);
    const int    r       = tid >> 6;               // OTILE == 64
    const float* xr      = tile + r * IN_DIM;
    float*       orow    = out + (size_t)(row0 + r) * OUT_DIM;
    const int    o_begin = blockIdx.x * OUT_PER_BLOCK;

    // First tile's parameters load concurrently with the async DMA
    // (LOADcnt and ASYNCcnt are independent counters).
    OutParams nxt = load_params(idx, w, theta, s_raw, o_begin + o_local);

    // Now drain the DMA and make the LDS tile visible workgroup-wide.
    asm volatile("s_wait_asynccnt 0x0" ::: "memory");
    __syncthreads();

    for (int ot = 0; ot < OUT_PER_BLOCK; ot += OTILE) {
        const int o = o_begin + ot + o_local;
        const OutParams p = nxt;

        // Warm WGP$ for the parameter stream two tiles ahead.
        if (ot + 2 * OTILE < OUT_PER_BLOCK) {
            __builtin_prefetch(idx + (size_t)(o + 2 * OTILE) * FAN_IN, 0, 0);
            __builtin_prefetch(w   + (size_t)(o + 2 * OTILE) * FAN_IN, 0, 0);
        }
        // Issue next tile's parameter loads before this tile's compute so the
        // vmem latency overlaps the gather + transcendental chain.
        if (ot + OTILE < OUT_PER_BLOCK) {
            nxt = load_params(idx, w, theta, s_raw, o + OTILE);
        }

        // 8 LDS gathers + FMA chain
        float z = fmaf(xr[p.i0.x], p.w0.x,
                  fmaf(xr[p.i0.y], p.w0.y,
                  fmaf(xr[p.i0.z], p.w0.z,
                  fmaf(xr[p.i0.w], p.w0.w,
                  fmaf(xr[p.i1.x], p.w1.x,
                  fmaf(xr[p.i1.y], p.w1.y,
                  fmaf(xr[p.i1.z], p.w1.z,
                       xr[p.i1.w] * p.w1.w)))))));
        z -= p.th;

        // stable softplus + sigmoid epilogue
        const float sp = fmaxf(p.sr, 0.0f) + log1pf(__expf(-fabsf(p.sr)));
        const float s  = sp + 1e-6f;
        orow[o] = 1.0f / (1.0f + __expf(-s * z));
    }
}

extern "C" void kernel_launch(void* const* d_in, const int* in_sizes, int n_in,
                              void* d_out, int out_size, void* d_ws, size_t ws_size,
                              hipStream_t stream) {
    (void)in_sizes; (void)n_in; (void)d_ws; (void)ws_size; (void)out_size;

    const float* x     = (const float*)d_in[0];
    const int*   idx   = (const int*)  d_in[1];   // jnp.int64 canonicalizes to i32
    const float* w     = (const float*)d_in[2];
    const float* theta = (const float*)d_in[3];
    const float* s_raw = (const float*)d_in[4];
    float*       out   = (float*)d_out;

    dim3 grid(OSPLIT, B_DIM / ROWS);              // 4 x 256 = 1024 workgroups
    dim3 block(THREADS);
    wtg_kernel<<<grid, block, 0, stream>>>(x, idx, w, theta, s_raw, out);
}